// GCNConv_76012331205022
// MI455X (gfx1250) — compile-verified
//
#include <hip/hip_runtime.h>
#include <hip/hip_bf16.h>

// GCNConv: out = Ddst^-1/2 * A * (Dsrc^-1/2 * X * W) + b
// N=100000 nodes, E=1600000 edges, DIN=DOUT=32, all fp32.
//
// WMMA: V_WMMA_F32_16X16X4_F32 (fp32 in/out -> no precision loss vs reference).

typedef float v2f __attribute__((ext_vector_type(2)));
typedef float v8f __attribute__((ext_vector_type(8)));

#define DIN  32
#define DOUT 32

// ---------------------------------------------------------------- zero
__global__ void gcn_zero_kernel(float* __restrict__ out,
                                float* __restrict__ deg, // 2*N floats (src then dst)
                                int N) {
    int i = blockIdx.x * blockDim.x + threadIdx.x;
    int total = N * DOUT;
    if (i < total) out[i] = 0.0f;
    if (i < 2 * N) deg[i] = 0.0f;
}

// ---------------------------------------------------------------- degrees
__global__ void gcn_degree_kernel(const int* __restrict__ ei, // [2,E]
                                  float* __restrict__ deg_src,
                                  float* __restrict__ deg_dst,
                                  int E) {
    int e = blockIdx.x * blockDim.x + threadIdx.x;
    if (e >= E) return;
    atomicAdd(&deg_src[ei[e]], 1.0f);
    atomicAdd(&deg_dst[ei[E + e]], 1.0f);
}

// ---------------------------------------------------------------- deg -> deg^-1/2 (in place, both arrays contiguous)
__global__ void gcn_norm_kernel(float* __restrict__ deg, int n2) {
    int i = blockIdx.x * blockDim.x + threadIdx.x;
    if (i >= n2) return;
    float d = deg[i];
    deg[i] = (d > 0.0f) ? (1.0f / sqrtf(d)) : 0.0f;
}

// ---------------------------------------------------------------- h = (x * norm_src) @ W  via V_WMMA_F32_16X16X4_F32
// One wave32 per 16-node tile. Two 16x16 output tiles (DOUT=32), 8 K-steps of 4.
// A layout (16x4 f32): lane half picks K-pair, vgpr=K%2, lanes 0-15 index M.
// B layout (4x16 f32): mirrored, lanes index N.
// C/D (16x16 f32, 8 vgprs): M = r + 8*(lane>=16), N = lane%16.
__global__ void gcn_transform_wmma_kernel(const float* __restrict__ x,
                                          const float* __restrict__ W,
                                          const float* __restrict__ norm_src,
                                          float* __restrict__ h,
                                          int N) {
    __shared__ float Ws[DIN * DOUT];
    for (int i = threadIdx.x; i < DIN * DOUT; i += blockDim.x) Ws[i] = W[i];
    __syncthreads();

    const int lane = threadIdx.x & 31;
    const int wave = threadIdx.x >> 5;
    const int tile = blockIdx.x * (blockDim.x >> 5) + wave;
    const int tiles = (N + 15) >> 4;
    if (tile >= tiles) return;          // wave-uniform: EXEC stays all-1s for WMMA

    const int m0   = tile << 4;
    const int half = lane >> 4;         // 0: lanes 0-15, 1: lanes 16-31
    const int ml   = lane & 15;

    int m = m0 + ml;
    const bool mv = (m < N);
    const int mc = mv ? m : (N - 1);    // clamp loads instead of predicating (EXEC must stay full)
    const float ns = mv ? norm_src[mc] : 0.0f;
    const float* xrow = x + (size_t)mc * DIN;

    v8f c0 = {};                         // n-cols 0..15
    v8f c1 = {};                         // n-cols 16..31

#pragma unroll
    for (int kk = 0; kk < DIN / 4; ++kk) {
        const int kb = kk * 4 + half * 2;   // this lane's two K values
        v2f a;
        a.x = xrow[kb + 0] * ns;
        a.y = xrow[kb + 1] * ns;
        v2f b0, b1;
        b0.x = Ws[(kb + 0) * DOUT + ml];
        b0.y = Ws[(kb + 1) * DOUT + ml];
        b1.x = Ws[(kb + 0) * DOUT + 16 + ml];
        b1.y = Ws[(kb + 1) * DOUT + 16 + ml];
        // (neg_a, A, neg_b, B, c_mod, C, reuse_a, reuse_b)
        c0 = __builtin_amdgcn_wmma_f32_16x16x4_f32(false, a, false, b0, (short)0, c0, false, false);
        c1 = __builtin_amdgcn_wmma_f32_16x16x4_f32(false, a, false, b1, (short)0, c1, false, false);
    }

#pragma unroll
    for (int r = 0; r < 8; ++r) {
        const int mr = m0 + r + half * 8;
        if (mr < N) {                    // divergence only after all WMMAs
            h[(size_t)mr * DOUT + ml]      = c0[r];
            h[(size_t)mr * DOUT + 16 + ml] = c1[r];
        }
    }
}

// ---------------------------------------------------------------- edge scatter: wave per edge, lane = feature
__global__ void gcn_scatter_kernel(const float* __restrict__ h,
                                   const int* __restrict__ ei, // [2,E]
                                   float* __restrict__ out,
                                   int E) {
    int gid = blockIdx.x * blockDim.x + threadIdx.x;
    int e = gid >> 5;
    int lane = gid & 31;
    if (e >= E) return;
    int s = ei[e];
    int d = ei[E + e];
    float v = h[(size_t)s * DOUT + lane];        // coalesced 128B gather
    atomicAdd(&out[(size_t)d * DOUT + lane], v); // global_atomic_add_f32
}

// ---------------------------------------------------------------- out = out * norm_dst[row] + b[col]  (in place)
__global__ void gcn_finalize_kernel(float* __restrict__ out,
                                    const float* __restrict__ norm_dst,
                                    const float* __restrict__ b,
                                    int N) {
    int i = blockIdx.x * blockDim.x + threadIdx.x;
    if (i >= N * DOUT) return;
    int row = i >> 5;
    int col = i & 31;
    out[i] = out[i] * norm_dst[row] + b[col];
}

extern "C" void kernel_launch(void* const* d_in, const int* in_sizes, int n_in,
                              void* d_out, int out_size, void* d_ws, size_t ws_size,
                              hipStream_t stream) {
    const float* x  = (const float*)d_in[0];   // [N, 32]
    const int*   ei = (const int*)d_in[1];     // [2, E]
    const float* W  = (const float*)d_in[2];   // [32, 32]
    const float* b  = (const float*)d_in[3];   // [32]
    float* out = (float*)d_out;                // [N, 32]

    const int N = in_sizes[0] / DIN;
    const int E = in_sizes[1] / 2;

    // workspace layout: deg_src[N] | deg_dst[N] | h[N*32]  (~13.6 MB)
    float* deg_src = (float*)d_ws;
    float* deg_dst = deg_src + N;
    float* h       = deg_dst + N;

    const int T = 256;

    // 1. zero accumulators (harness does not re-zero between replays)
    {
        int total = N * DOUT;                   // covers 2*N as well
        gcn_zero_kernel<<<(total + T - 1) / T, T, 0, stream>>>(out, deg_src, N);
    }
    // 2. degrees
    gcn_degree_kernel<<<(E + T - 1) / T, T, 0, stream>>>(ei, deg_src, deg_dst, E);
    // 3. deg -> deg^-1/2 (both arrays, contiguous 2N)
    gcn_norm_kernel<<<(2 * N + T - 1) / T, T, 0, stream>>>(deg_src, 2 * N);
    // 4. feature transform via f32 WMMA: one wave per 16-row tile, 8 waves/block
    {
        int tiles = (N + 15) / 16;
        int wavesPerBlock = T / 32;
        gcn_transform_wmma_kernel<<<(tiles + wavesPerBlock - 1) / wavesPerBlock, T, 0, stream>>>(
            x, W, deg_src /*norm_src*/, h, N);
    }
    // 5. sparse aggregation: warp per edge, lane per feature
    {
        long long threads = (long long)E * 32;
        gcn_scatter_kernel<<<(int)((threads + T - 1) / T), T, 0, stream>>>(h, ei, out, E);
    }
    // 6. finalize: scale by norm_dst, add bias
    gcn_finalize_kernel<<<(N * DOUT + T - 1) / T, T, 0, stream>>>(out, deg_dst /*norm_dst*/, b, N);
}